// PrimalDual_1494648619549
// MI455X (gfx1250) — compile-verified
//
#include <hip/hip_runtime.h>

// ---------------- problem constants ----------------
#define HH   768
#define WW   768
#define LL   8
#define PRJ  36
#define NPIX (HH * WW)                       // 589824
#define NLVL ((size_t)NPIX * LL)             // 4718592
#define NPRJ ((size_t)NPIX * PRJ)            // 21233664

#define LMBDA   0.1f
#define NU      0.01f
#define SIGMAP  (1.0f / 11.0f)               // 1/(3+L)
#define SIGMAS  1.0f
#define TAUU    (1.0f / 6.0f)
#define TAUMU   (1.0f / 11.0f)               // 1/(2 + PROJ/4)

typedef __attribute__((ext_vector_type(2))) float v2f;
typedef __attribute__((ext_vector_type(4))) float v4f;
typedef __attribute__((ext_vector_type(8))) float v8f;

// Per-z membership bitmask over the 36 combinations K=(k1,k2), k1<=k2:
// bit p of MTAB[z] == 1  iff  k1(p) <= z <= k2(p).  z in 8..15 -> 0 (zero
// rows of the 16x16 WMMA tile).
struct MaskTab { unsigned long long m[16]; };
static constexpr MaskTab make_masks() {
  MaskTab t{};
  for (int z = 0; z < 16; ++z) {
    unsigned long long m = 0;
    int K = 0;
    for (int a = 0; a < LL; ++a)
      for (int b = a; b < LL; ++b) {
        if (a <= z && z <= b) m |= (1ull << K);
        ++K;
      }
    t.m[z] = m;
  }
  return t;
}
static __device__ __constant__ MaskTab MT = make_masks();

// Cubic-root projection onto the parabola (elementwise epilogue of _parabola).
__device__ __forceinline__ void parab_update(float u1, float u2, float u3,
                                             float quad, float& o1, float& o2,
                                             float& o3) {
  float nrm2 = u1 * u1 + u2 * u2;
  float Bv = 0.25f * nrm2 - quad;
  bool mask = u3 < Bv;
  float y = u3 + quad;
  float nrm = sqrtf(nrm2);
  float a = 0.5f * nrm;
  float b = (2.0f / 3.0f) * (1.0f - 0.5f * y);
  float sb = sqrtf(fmaxf(-b, 0.0f));
  float sb3 = sb * sb * sb;
  float d = (b < 0.0f) ? (a - sb3) * (a + sb3) : (a * a + b * b * b);
  float sd = sqrtf(fmaxf(d, 0.0f));
  float c = cbrtf(a + sd);
  float c_safe = (c == 0.0f) ? 1.0f : c;
  float sb3s = (sb > 0.0f) ? sb3 : 1.0f;
  float cosarg = fminf(fmaxf(a / sb3s, -1.0f), 1.0f);
  float v_neg = 2.0f * sb * cosf(acosf(cosarg) * (1.0f / 3.0f));
  float v = ((d >= 0.0f) && (c == 0.0f))
                ? 0.0f
                : ((d < 0.0f) ? v_neg : (c - b / c_safe));
  float nrm_safe = (nrm == 0.0f) ? 1.0f : nrm;
  float scale = 2.0f * v / nrm_safe;
  float p1n = mask ? ((nrm == 0.0f) ? 0.0f : scale * u1) : u1;
  float p2n = mask ? ((nrm == 0.0f) ? 0.0f : scale * u2) : u2;
  o1 = p1n;
  o2 = p2n;
  o3 = mask ? (0.25f * (p1n * p1n + p2n * p2n) - quad) : u3;
}

// ---------------------------------------------------------------------------
// Kernel 1: parabola.  Block = 128 threads = 4 waves; each wave owns 32 pixels.
// mu1sum/mu2sum computed with V_WMMA_F32_16X16X4_F32:
//   A (16x4)  = membership M rows z (z>=8 rows are zero), K-chunk of proj dim,
//               built branch-free from the per-lane bitmask (uniform EXEC
//               through the whole WMMA section, as the ISA requires)
//   B (4x16)  = mu values for 16 pixels (gathered from LDS stage)
//   C (16x16) accumulated over 9 K-steps;  D[r][lane] = (z=r+8*(lane>=16),
//   pixel=lane&15).  One shuffle per register moves tile-1 results to lanes
//   16..31 so the epilogue runs with lane == pixel (coalesced f32x4 I/O).
// ---------------------------------------------------------------------------
__global__ __launch_bounds__(128) void k_parabola(
    const float* __restrict__ ubar, const float* __restrict__ p1,
    const float* __restrict__ p2, const float* __restrict__ p3,
    const float* __restrict__ mu1, const float* __restrict__ mu2,
    const float* __restrict__ f, float* __restrict__ p1o,
    float* __restrict__ p2o, float* __restrict__ p3o) {
  __shared__ float lm1[128 * PRJ];
  __shared__ float lm2[128 * PRJ];
  const int tid = threadIdx.x;
  const int blockPix = (int)blockIdx.x * 128;

  // Coalesced stage of this block's mu tiles into LDS (uniform trip count).
  for (int j = tid; j < 128 * PRJ; j += 128) {
    lm1[j] = mu1[(size_t)blockPix * PRJ + j];
    lm2[j] = mu2[(size_t)blockPix * PRJ + j];
  }
  __syncthreads();

  const int lane = tid & 31;
  const int wave = tid >> 5;
  const int n = lane & 15;   // B column / pixel-in-tile
  const int kh = lane >> 4;  // K half-select for A/B VGPR layout
  const int z16 = lane & 15; // A row (z)

  // Per-lane membership bits, pre-shifted so bit (p0+v) is A[v] for this lane.
  const unsigned long long amask = MT.m[z16] >> (2 * kh);

  // LDS base for this lane's B gathers (v=0 element; v=1 is +1).
  const int b0base = (wave * 32 + n) * PRJ + 2 * kh;       // tile 0: pixels 0-15
  const int b1base = (wave * 32 + 16 + n) * PRJ + 2 * kh;  // tile 1: pixels 16-31

  v8f a10 = {}, a11 = {}, a20 = {}, a21 = {};
#pragma unroll
  for (int p0 = 0; p0 < PRJ; p0 += 4) {
    v2f a;  // A[v][lane]: K = v + 2*kh, M = z16 (bit extract, branch-free)
    a.x = (float)(unsigned int)((amask >> p0) & 1ull);
    a.y = (float)(unsigned int)((amask >> (p0 + 1)) & 1ull);
    v2f b;
    b.x = lm1[b0base + p0];
    b.y = lm1[b0base + p0 + 1];
    a10 = __builtin_amdgcn_wmma_f32_16x16x4_f32(false, a, false, b, (short)0,
                                                a10, false, false);
    b.x = lm1[b1base + p0];
    b.y = lm1[b1base + p0 + 1];
    a11 = __builtin_amdgcn_wmma_f32_16x16x4_f32(false, a, false, b, (short)0,
                                                a11, false, false);
    b.x = lm2[b0base + p0];
    b.y = lm2[b0base + p0 + 1];
    a20 = __builtin_amdgcn_wmma_f32_16x16x4_f32(false, a, false, b, (short)0,
                                                a20, false, false);
    b.x = lm2[b1base + p0];
    b.y = lm2[b1base + p0 + 1];
    a21 = __builtin_amdgcn_wmma_f32_16x16x4_f32(false, a, false, b, (short)0,
                                                a21, false, false);
  }

  // Redistribute: lane l<16 -> tile0 D[r][l]; lane l>=16 -> tile1 D[r][l-16].
  float ms1[8], ms2[8];
#pragma unroll
  for (int r = 0; r < 8; ++r) {
    float t1 = __shfl(a11[r], n, 32);
    float t2 = __shfl(a21[r], n, 32);
    ms1[r] = (lane < 16) ? a10[r] : t1;
    ms2[r] = (lane < 16) ? a20[r] : t2;
  }

  // ---- per-pixel elementwise epilogue (lane == pixel) ----
  const int pix = blockPix + wave * 32 + lane;
  const int h = pix / WW;
  const int w = pix % WW;
  const size_t base = (size_t)pix * LL;

  v4f ua = *(const v4f*)(ubar + base);
  v4f ubv = *(const v4f*)(ubar + base + 4);
  float ub[8] = {ua.x, ua.y, ua.z, ua.w, ubv.x, ubv.y, ubv.z, ubv.w};

  float duh[8], duw[8];
  if (h < HH - 1) {
    v4f n0 = *(const v4f*)(ubar + base + (size_t)WW * LL);
    v4f n1 = *(const v4f*)(ubar + base + (size_t)WW * LL + 4);
    float t[8] = {n0.x, n0.y, n0.z, n0.w, n1.x, n1.y, n1.z, n1.w};
#pragma unroll
    for (int z = 0; z < 8; ++z) duh[z] = t[z] - ub[z];
  } else {
#pragma unroll
    for (int z = 0; z < 8; ++z) duh[z] = 0.0f;
  }
  if (w < WW - 1) {
    v4f n0 = *(const v4f*)(ubar + base + LL);
    v4f n1 = *(const v4f*)(ubar + base + LL + 4);
    float t[8] = {n0.x, n0.y, n0.z, n0.w, n1.x, n1.y, n1.z, n1.w};
#pragma unroll
    for (int z = 0; z < 8; ++z) duw[z] = t[z] - ub[z];
  } else {
#pragma unroll
    for (int z = 0; z < 8; ++z) duw[z] = 0.0f;
  }

  v4f q0 = *(const v4f*)(p1 + base);
  v4f q1 = *(const v4f*)(p1 + base + 4);
  float p1v[8] = {q0.x, q0.y, q0.z, q0.w, q1.x, q1.y, q1.z, q1.w};
  q0 = *(const v4f*)(p2 + base);
  q1 = *(const v4f*)(p2 + base + 4);
  float p2v[8] = {q0.x, q0.y, q0.z, q0.w, q1.x, q1.y, q1.z, q1.w};
  q0 = *(const v4f*)(p3 + base);
  q1 = *(const v4f*)(p3 + base + 4);
  float p3v[8] = {q0.x, q0.y, q0.z, q0.w, q1.x, q1.y, q1.z, q1.w};
  const float fv = f[pix];

  float r1[8], r2[8], r3[8];
#pragma unroll
  for (int z = 0; z < 8; ++z) {
    float u1 = p1v[z] + SIGMAP * (duh[z] + ms1[z]);
    float u2 = p2v[z] + SIGMAP * (duw[z] + ms2[z]);
    float du3 = (z < 7) ? (ub[z + 1] - ub[z]) : 0.0f;
    float u3 = p3v[z] + SIGMAP * du3;
    float df = (float)(z + 1) * 0.125f - fv;
    float quad = LMBDA * df * df;
    parab_update(u1, u2, u3, quad, r1[z], r2[z], r3[z]);
  }
  v4f o;
  o.x = r1[0]; o.y = r1[1]; o.z = r1[2]; o.w = r1[3];
  *(v4f*)(p1o + base) = o;
  o.x = r1[4]; o.y = r1[5]; o.z = r1[6]; o.w = r1[7];
  *(v4f*)(p1o + base + 4) = o;
  o.x = r2[0]; o.y = r2[1]; o.z = r2[2]; o.w = r2[3];
  *(v4f*)(p2o + base) = o;
  o.x = r2[4]; o.y = r2[5]; o.z = r2[6]; o.w = r2[7];
  *(v4f*)(p2o + base + 4) = o;
  o.x = r3[0]; o.y = r3[1]; o.z = r3[2]; o.w = r3[3];
  *(v4f*)(p3o + base) = o;
  o.x = r3[4]; o.y = r3[5]; o.z = r3[6]; o.w = r3[7];
  *(v4f*)(p3o + base + 4) = o;
}

// ---------------------------------------------------------------------------
// Kernel 2: fused l2projection + mu update.  One thread per (pixel, comb).
// t1/t2 (reverse membership einsum) = sum of p over levels k1..k2.
// ---------------------------------------------------------------------------
__global__ __launch_bounds__(256) void k_projmu(
    const float* __restrict__ s1, const float* __restrict__ s2,
    const float* __restrict__ mubar1, const float* __restrict__ mubar2,
    const float* __restrict__ mu1, const float* __restrict__ mu2,
    const float* __restrict__ p1, const float* __restrict__ p2,
    float* __restrict__ s1o, float* __restrict__ s2o, float* __restrict__ mu1o,
    float* __restrict__ mu2o, float* __restrict__ mubar1o,
    float* __restrict__ mubar2o) {
  size_t i = (size_t)blockIdx.x * blockDim.x + threadIdx.x;  // exact multiple
  int c = (int)(i % PRJ);
  size_t pix = i / PRJ;

  // decode combination index -> (k1, k2)
  int k1 = 0, rem = c;
  while (rem >= LL - k1) { rem -= LL - k1; ++k1; }
  int k2 = k1 + rem;

  float t1 = 0.0f, t2 = 0.0f;
  const size_t pb = pix * LL;
  for (int z = k1; z <= k2; ++z) {
    t1 += p1[pb + z];
    t2 += p2[pb + z];
  }

  float m1 = s1[i] - SIGMAS * mubar1[i];
  float m2 = s2[i] - SIGMAS * mubar2[i];
  float nrm = sqrtf(m1 * m1 + m2 * m2);
  bool mk = nrm > NU;
  float inv = mk ? (NU / nrm) : 1.0f;
  float s1n = m1 * inv;
  float s2n = m2 * inv;

  float mu1v = mu1[i];
  float mu2v = mu2[i];
  float mu1n = mu1v + TAUMU * (s1n - t1);
  float mu2n = mu2v + TAUMU * (s2n - t2);

  s1o[i] = s1n;
  s2o[i] = s2n;
  mu1o[i] = mu1n;
  mu2o[i] = mu2n;
  mubar1o[i] = 2.0f * mu1n - mu1v;
  mubar2o[i] = 2.0f * mu2n - mu2v;
}

// ---------------------------------------------------------------------------
// Kernel 3: clipping.  One thread per (pixel, level) element; divergence via
// neighbor gathers (h-1: -W*L, w-1: -L, z-1: -1).
// ---------------------------------------------------------------------------
__global__ __launch_bounds__(256) void k_clipping(
    const float* __restrict__ u, const float* __restrict__ p1,
    const float* __restrict__ p2, const float* __restrict__ p3,
    float* __restrict__ uo, float* __restrict__ ubaro, int writeUbar) {
  size_t i = (size_t)blockIdx.x * blockDim.x + threadIdx.x;  // exact multiple
  int z = (int)(i & (LL - 1));
  size_t pw = i >> 3;
  int w = (int)(pw % WW);
  int h = (int)(pw / WW);

  // speculative read-ahead of the streaming operand (global_prefetch_b8)
  if (i + 32768 < NLVL) __builtin_prefetch(p1 + i + 32768, 0, 0);

  float d1 = ((h < HH - 1) ? p1[i] : 0.0f) -
             ((h > 0) ? p1[i - (size_t)WW * LL] : 0.0f);
  float d2 = ((w < WW - 1) ? p2[i] : 0.0f) - ((w > 0) ? p2[i - LL] : 0.0f);
  float d3 = ((z < LL - 1) ? p3[i] : 0.0f) - ((z > 0) ? p3[i - 1] : 0.0f);

  float uv = u[i];
  float un = fminf(fmaxf(uv + TAUU * (d1 + d2 + d3), 0.0f), 1.0f);
  uo[i] = un;
  if (writeUbar) ubaro[i] = 2.0f * un - uv;
}

// ---------------------------------------------------------------------------
extern "C" void kernel_launch(void* const* d_in, const int* in_sizes, int n_in,
                              void* d_out, int out_size, void* d_ws,
                              size_t ws_size, hipStream_t stream) {
  (void)in_sizes; (void)n_in; (void)out_size;

  const float* in_u      = (const float*)d_in[0];
  const float* in_ubar   = (const float*)d_in[1];
  const float* in_p1     = (const float*)d_in[2];
  const float* in_p2     = (const float*)d_in[3];
  const float* in_p3     = (const float*)d_in[4];
  const float* in_s1     = (const float*)d_in[5];
  const float* in_s2     = (const float*)d_in[6];
  const float* in_mu1    = (const float*)d_in[7];
  const float* in_mu2    = (const float*)d_in[8];
  const float* in_mubar1 = (const float*)d_in[9];
  const float* in_mubar2 = (const float*)d_in[10];
  const float* in_f      = (const float*)d_in[11];
  float* uo = (float*)d_out;

  // workspace layout: 4 level fields + 6 proj fields (u lives in d_out)
  const size_t need = (4 * NLVL + 6 * NPRJ) * sizeof(float);
  if (ws_size < need) return;  // cannot run without scratch
  float* ws = (float*)d_ws;
  float* ubar_ws   = ws;
  float* p1_ws     = ubar_ws + NLVL;
  float* p2_ws     = p1_ws + NLVL;
  float* p3_ws     = p2_ws + NLVL;
  float* s1_ws     = p3_ws + NLVL;
  float* s2_ws     = s1_ws + NPRJ;
  float* mu1_ws    = s2_ws + NPRJ;
  float* mu2_ws    = mu1_ws + NPRJ;
  float* mubar1_ws = mu2_ws + NPRJ;
  float* mubar2_ws = mubar1_ws + NPRJ;

  const dim3 bPar(128), bEl(256);
  const dim3 gPar(NPIX / 128);              // 4608, exact
  const dim3 gProj((unsigned)(NPRJ / 256)); // 82944, exact
  const dim3 gLvl((unsigned)(NLVL / 256));  // 18432, exact

  // ---- iteration 1: read d_in, write ws / d_out ----
  k_parabola<<<gPar, bPar, 0, stream>>>(in_ubar, in_p1, in_p2, in_p3, in_mu1,
                                        in_mu2, in_f, p1_ws, p2_ws, p3_ws);
  k_projmu<<<gProj, bEl, 0, stream>>>(in_s1, in_s2, in_mubar1, in_mubar2,
                                      in_mu1, in_mu2, p1_ws, p2_ws, s1_ws,
                                      s2_ws, mu1_ws, mu2_ws, mubar1_ws,
                                      mubar2_ws);
  k_clipping<<<gLvl, bEl, 0, stream>>>(in_u, p1_ws, p2_ws, p3_ws, uo, ubar_ws,
                                       1);

  // ---- iteration 2: in-place in ws; final u -> d_out ----
  k_parabola<<<gPar, bPar, 0, stream>>>(ubar_ws, p1_ws, p2_ws, p3_ws, mu1_ws,
                                        mu2_ws, in_f, p1_ws, p2_ws, p3_ws);
  k_projmu<<<gProj, bEl, 0, stream>>>(s1_ws, s2_ws, mubar1_ws, mubar2_ws,
                                      mu1_ws, mu2_ws, p1_ws, p2_ws, s1_ws,
                                      s2_ws, mu1_ws, mu2_ws, mubar1_ws,
                                      mubar2_ws);
  k_clipping<<<gLvl, bEl, 0, stream>>>(uo, p1_ws, p2_ws, p3_ws, uo, ubar_ws,
                                       0);
}